// RaggedNorm_41781441855970
// MI455X (gfx1250) — compile-verified
//
#include <hip/hip_runtime.h>
#include <math.h>

typedef __attribute__((ext_vector_type(2))) float v2f;
typedef __attribute__((ext_vector_type(4))) float v4f;
typedef __attribute__((ext_vector_type(8))) float v8f;

#define N_ROWS 524288
#define N_SEG  128
#define N_FEAT 256
#define EPS    1e-5f

// ---------------------------------------------------------------------------
// Kernel 1: serial prefix sum of segment lengths -> offsets[0..N_SEG]
// (lengths live only on device; grid shape must be data-independent)
// ---------------------------------------------------------------------------
__global__ void k_prefix(const int* __restrict__ lengths, int* __restrict__ offs) {
    if (blockIdx.x == 0 && threadIdx.x == 0) {
        int acc = 0;
        offs[0] = 0;
        for (int i = 0; i < N_SEG; ++i) {
            acc += lengths[i];
            offs[i + 1] = acc;
        }
    }
}

// ---------------------------------------------------------------------------
// Kernel 2: zero the sum / sumsq accumulators (ws is poisoned by the harness)
// ---------------------------------------------------------------------------
__global__ void k_zero(float* __restrict__ sums, float* __restrict__ sumsq) {
    int i = blockIdx.x * blockDim.x + threadIdx.x;   // 128*256 = 32768 threads
    sums[i]  = 0.0f;
    sumsq[i] = 0.0f;
}

// ---------------------------------------------------------------------------
// Kernel 3: per-(segment,feature) sum and sum-of-squares via WMMA reduction.
//   D = A(16x4) * ones(4x16) + C  =>  D[m][n] = sum_k A[m][k]  (all n equal)
//   A layout (ISA 7.12.2, 32-bit A 16x4): lanes 0-15: M=lane, V0=K0,V1=K1;
//                                         lanes 16-31: M=lane-16, V0=K2,V1=K3.
//   C/D layout: lane 0 vgpr v = (M=v, N=0); lane 16 vgpr v = (M=v+8, N=0).
// Each wave owns 2 groups of 16 features. Main loop: 16 rows / 4 WMMA-pairs
// per iteration with unconditional loads off one base address (immediate
// offsets, clause-able); only the <=15-row segment tail takes the masked path.
// ---------------------------------------------------------------------------
__global__ void __launch_bounds__(256)
k_stats(const float* __restrict__ x, const int* __restrict__ offs,
        float* __restrict__ sums, float* __restrict__ sumsq) {
    const int ROWS_PER_BLOCK = 512;
    const int rBeg = blockIdx.x * ROWS_PER_BLOCK;
    const int rEnd = rBeg + ROWS_PER_BLOCK;

    const int tid  = threadIdx.x;
    const int wave = tid >> 5;
    const int lane = tid & 31;
    const int l16  = lane & 15;   // M within half / feature within group
    const int hi   = lane >> 4;   // 0: K0,K1   1: K2,K3

    v2f ones;
    ones[0] = 1.0f;
    ones[1] = 1.0f;

    for (int fi = 0; fi < 2; ++fi) {
        const int fg    = wave + fi * 8;        // feature group 0..15
        const int fLoad = fg * 16 + l16;        // feature column this lane loads

        // locate segment containing rBeg (offsets are uniform; tiny scan)
        int s = 0;
        while (offs[s + 1] <= rBeg) ++s;

        int r = rBeg;
        while (r < rEnd) {
            const int segEnd = offs[s + 1];
            const int stop   = segEnd < rEnd ? segEnd : rEnd;

            v8f accS = {0.f, 0.f, 0.f, 0.f, 0.f, 0.f, 0.f, 0.f};
            v8f accQ = {0.f, 0.f, 0.f, 0.f, 0.f, 0.f, 0.f, 0.f};

            int rr = r;
            // ---- main loop: 16 rows, unconditional loads, 4 WMMA pairs ----
            for (; rr + 16 <= stop; rr += 16) {
                const float* p = x + (size_t)(rr + 2 * hi) * N_FEAT + fLoad;
                #pragma unroll
                for (int u = 0; u < 4; ++u) {
                    float a0 = p[u * 4 * N_FEAT];              // row rr+4u+2hi
                    float a1 = p[u * 4 * N_FEAT + N_FEAT];     // row rr+4u+2hi+1
                    v2f A; A[0] = a0; A[1] = a1;
                    accS = __builtin_amdgcn_wmma_f32_16x16x4_f32(
                        false, A, false, ones, (short)0, accS, false, false);
                    v2f Q; Q[0] = a0 * a0; Q[1] = a1 * a1;
                    accQ = __builtin_amdgcn_wmma_f32_16x16x4_f32(
                        false, Q, false, ones, (short)0, accQ, false, false);
                }
            }
            // ---- tail: <=15 rows left in this segment chunk ----
            for (; rr < stop; rr += 4) {
                const int r0 = rr + 2 * hi;
                const int r1 = r0 + 1;
                float a0 = (r0 < stop) ? x[(size_t)r0 * N_FEAT + fLoad] : 0.0f;
                float a1 = (r1 < stop) ? x[(size_t)r1 * N_FEAT + fLoad] : 0.0f;
                v2f A; A[0] = a0; A[1] = a1;
                accS = __builtin_amdgcn_wmma_f32_16x16x4_f32(
                    false, A, false, ones, (short)0, accS, false, false);
                v2f Q; Q[0] = a0 * a0; Q[1] = a1 * a1;
                accQ = __builtin_amdgcn_wmma_f32_16x16x4_f32(
                    false, Q, false, ones, (short)0, accQ, false, false);
            }

            // lane 0 holds features fg*16+0..7 in vgprs 0..7 (N=0 column),
            // lane 16 holds features fg*16+8..15
            if (l16 == 0) {
                const int fBase = fg * 16 + hi * 8;
                #pragma unroll
                for (int v = 0; v < 8; ++v) {
                    atomicAdd(&sums [s * N_FEAT + fBase + v], accS[v]);
                    atomicAdd(&sumsq[s * N_FEAT + fBase + v], accQ[v]);
                }
            }
            r = stop;
            ++s;
        }
    }
}

// ---------------------------------------------------------------------------
// Kernel 4: fold stats + weight/bias into per-(seg,feat) scale/shift, in place
//   a = w * rsqrt(var + eps),  b = bias - mean * a
// ---------------------------------------------------------------------------
__global__ void k_final(const int* __restrict__ offs,
                        float* __restrict__ sums, float* __restrict__ sumsq,
                        const float* __restrict__ w, const float* __restrict__ bias) {
    const int s = blockIdx.x;
    const int f = threadIdx.x;
    const int idx = s * N_FEAT + f;
    const float cnt  = (float)(offs[s + 1] - offs[s]);
    const float mean = sums[idx] / cnt;
    float var = sumsq[idx] / cnt - mean * mean;
    var = var < 0.0f ? 0.0f : var;
    const float inv = 1.0f / sqrtf(var + EPS);
    const float a = w[f] * inv;
    const float b = bias[f] - mean * a;
    sums[idx]  = a;
    sumsq[idx] = b;
}

// ---------------------------------------------------------------------------
// Kernel 5: streaming normalize, float4 per thread (b128 ld/st, NT hints).
// 64 threads cover one 1KB row; block processes 4 rows per iteration. Row is
// uniform within a wave, so segment reloads are rare and cheap.
// ---------------------------------------------------------------------------
__global__ void __launch_bounds__(256)
k_norm(const float* __restrict__ x, const int* __restrict__ offs,
       const float* __restrict__ As, const float* __restrict__ Bs,
       float* __restrict__ out) {
    const int ROWS_PER_BLOCK = 256;
    const int rowBase = blockIdx.x * ROWS_PER_BLOCK;
    const int rBeg = rowBase + (threadIdx.x >> 6);     // this thread's first row
    const int rEnd = rowBase + ROWS_PER_BLOCK;
    const int fb   = (threadIdx.x & 63) * 4;           // 4 features per thread

    int s = 0;
    while (offs[s + 1] <= rBeg) ++s;
    int segEnd = offs[s + 1];
    v4f a = *(const v4f*)(As + s * N_FEAT + fb);
    v4f b = *(const v4f*)(Bs + s * N_FEAT + fb);

    for (int r = rBeg; r < rEnd; r += 4) {
        while (r >= segEnd) {            // lengths >= 1, advances correctly
            ++s;
            segEnd = offs[s + 1];
            a = *(const v4f*)(As + s * N_FEAT + fb);
            b = *(const v4f*)(Bs + s * N_FEAT + fb);
        }
        const size_t idx = (size_t)r * N_FEAT + fb;
        if (r + 16 < rEnd)
            __builtin_prefetch(&x[idx + 16 * N_FEAT], 0, 0);  // global_prefetch_b8
        v4f v = __builtin_nontemporal_load((const v4f*)(x + idx));
        v4f o;
        o[0] = fmaf(v[0], a[0], b[0]);
        o[1] = fmaf(v[1], a[1], b[1]);
        o[2] = fmaf(v[2], a[2], b[2]);
        o[3] = fmaf(v[3], a[3], b[3]);
        __builtin_nontemporal_store(o, (v4f*)(out + idx));
    }
}

// ---------------------------------------------------------------------------
extern "C" void kernel_launch(void* const* d_in, const int* in_sizes, int n_in,
                              void* d_out, int out_size, void* d_ws, size_t ws_size,
                              hipStream_t stream) {
    const float* x       = (const float*)d_in[0];
    const int*   lengths = (const int*)  d_in[1];
    const float* weight  = (const float*)d_in[2];
    const float* bias    = (const float*)d_in[3];
    float* out = (float*)d_out;

    char*  ws    = (char*)d_ws;
    int*   offs  = (int*)ws;                      // 129 ints (pad to 512 B)
    float* sums  = (float*)(ws + 512);            // 128*256 floats
    float* sumsq = sums + N_SEG * N_FEAT;         // 128*256 floats

    k_prefix<<<1, 1, 0, stream>>>(lengths, offs);
    k_zero  <<<(N_SEG * N_FEAT) / 256, 256, 0, stream>>>(sums, sumsq);
    k_stats <<<N_ROWS / 512, 256, 0, stream>>>(x, offs, sums, sumsq);
    k_final <<<N_SEG, N_FEAT, 0, stream>>>(offs, sums, sumsq, weight, bias);
    k_norm  <<<N_ROWS / 256, 256, 0, stream>>>(x, offs, sums, sumsq, out);
}